// DigitCaps_1735166788196
// MI455X (gfx1250) — compile-verified
//
#include <hip/hip_runtime.h>
#include <hip/hip_bf16.h>
#include <math.h>

// DigitCaps dynamic routing, fully fused for MI455X (gfx1250, wave32).
// u_hat (377 MB) is NEVER materialized: each pass recomputes the per-route
// 16x16x8 GEMM tiles with V_WMMA_F32_16X16X4_F32 (f32 exact class, matching
// the f32 reference). HBM traffic ~25 MB instead of ~2.3 GB.
//
// Sizes: B=512, R=1152, J=10, O=16, I=8.

#define BATCH 512
#define ROUTES 1152
#define NCAPS 10
#define OCH 16
#define ICH 8
#define RCHUNKS 8
#define RCHUNK_LEN (ROUTES / RCHUNKS)   // 144
#define BTILES (BATCH / 16)             // 32
#define SVOL (BATCH * NCAPS * OCH)      // 81920

typedef __attribute__((ext_vector_type(2))) float v2f;
typedef __attribute__((ext_vector_type(8))) float v8f;

__device__ __forceinline__ v8f wmma_k4(v2f a, v2f b, v8f c) {
  // 8 args: (neg_a, A, neg_b, B, c_mod, C, reuse_a, reuse_b)
  return __builtin_amdgcn_wmma_f32_16x16x4_f32(false, a, false, b,
                                               (short)0, c, false, false);
}

// ---------------------------------------------------------------- init b = 0
__global__ void init_b_kernel(float* __restrict__ b) {
  int i = blockIdx.x * blockDim.x + threadIdx.x;
  if (i < ROUTES * NCAPS) b[i] = 0.0f;
}

// -------------------------------------------- c[r,j] = softmax over r of b
__global__ void softmax_routes(const float* __restrict__ b,
                               float* __restrict__ c) {
  int j = blockIdx.x;           // 0..9
  int t = threadIdx.x;          // 0..255
  __shared__ float red[256];

  float mx = -__builtin_inff();
  for (int r = t; r < ROUTES; r += 256) mx = fmaxf(mx, b[r * NCAPS + j]);
  red[t] = mx; __syncthreads();
  for (int s = 128; s > 0; s >>= 1) {
    if (t < s) red[t] = fmaxf(red[t], red[t + s]);
    __syncthreads();
  }
  mx = red[0]; __syncthreads();

  float sum = 0.0f;
  for (int r = t; r < ROUTES; r += 256) sum += __expf(b[r * NCAPS + j] - mx);
  red[t] = sum; __syncthreads();
  for (int s = 128; s > 0; s >>= 1) {
    if (t < s) red[t] += red[t + s];
    __syncthreads();
  }
  float inv = 1.0f / red[0];
  for (int r = t; r < ROUTES; r += 256)
    c[r * NCAPS + j] = __expf(b[r * NCAPS + j] - mx) * inv;
}

// --- s_partial[rc][b,j,o] = sum_{r in chunk} c[r,j] * (W[r,j,o,:] . x[b,r,:])
// One wave per (batch_tile, j, r_chunk). 2 WMMAs (K=4 each) per route.
__global__ __launch_bounds__(32)
void compute_s_partial(const float* __restrict__ x,
                       const float* __restrict__ W,
                       const float* __restrict__ c,
                       float* __restrict__ s_partial) {
  int bid = blockIdx.x;
  int rc = bid % RCHUNKS;
  int j  = (bid / RCHUNKS) % NCAPS;
  int bt = bid / (RCHUNKS * NCAPS);
  int lane = threadIdx.x;
  int m  = lane & 15;    // A: batch row in tile; B: O column
  int hi = lane >> 4;    // K half selector
  int b0 = bt * 16;

  v8f acc = {};
  int r0 = rc * RCHUNK_LEN;
  for (int r = r0; r < r0 + RCHUNK_LEN; ++r) {
    const float* xp = x + ((size_t)(b0 + m) * ROUTES + r) * ICH;
    const float* wp = W + (((size_t)r * NCAPS + j) * OCH + m) * ICH;
    __builtin_prefetch(wp + NCAPS * OCH * ICH, 0, 0);  // next route's W row

    // A operand (16x4 f32): VGPR0 = K=2*hi, VGPR1 = K=2*hi+1
    v2f a0 = { xp[2 * hi + 0], xp[2 * hi + 1] };
    v2f a1 = { xp[4 + 2 * hi + 0], xp[4 + 2 * hi + 1] };
    // B operand (4x16 f32), column n=m shares c[r,j] (all 16 cols are one j)
    float cs = c[r * NCAPS + j];
    v2f bv0 = { cs * wp[2 * hi + 0], cs * wp[2 * hi + 1] };
    v2f bv1 = { cs * wp[4 + 2 * hi + 0], cs * wp[4 + 2 * hi + 1] };

    acc = wmma_k4(a0, bv0, acc);   // K = 0..3
    acc = wmma_k4(a1, bv1, acc);   // K = 4..7
  }

  // C/D layout: VGPR g holds element (M = g + 8*hi, N = m)
  float* sp = s_partial + (size_t)rc * SVOL;
  #pragma unroll
  for (int g = 0; g < 8; ++g) {
    int M = g + 8 * hi;
    sp[((size_t)(b0 + M) * NCAPS + j) * OCH + m] = acc[g];
  }
}

// -------------------- v = squash(sum of partials); v written into d_out
__global__ void reduce_squash(const float* __restrict__ s_partial,
                              float* __restrict__ v_out) {
  int i = blockIdx.x * blockDim.x + threadIdx.x;
  if (i >= SVOL) return;
  float s = 0.0f;
  #pragma unroll
  for (int rc = 0; rc < RCHUNKS; ++rc) s += s_partial[rc * SVOL + i];
  // squash elementwise: s^2*s / ((1+s^2)*|s|) = s*|s| / (1+s^2)
  float sq = s * s;
  v_out[i] = s * fabsf(s) / (1.0f + sq);
}

// ---- b[r,j] += mean_b( sum_o u_hat[b,r,j,o]*v[b,j,o] ), u_hat recomputed.
// One wave per (r,j); loops all 32 batch tiles -> deterministic order, and
// owns b[r,j] exclusively (no atomics).
__global__ __launch_bounds__(32)
void agreement_update_b(const float* __restrict__ x,
                        const float* __restrict__ W,
                        const float* __restrict__ v,
                        float* __restrict__ b) {
  int bid = blockIdx.x;        // r*NCAPS + j
  int j = bid % NCAPS;
  int r = bid / NCAPS;
  int lane = threadIdx.x;
  int m  = lane & 15;
  int hi = lane >> 4;

  const float* wp = W + (((size_t)r * NCAPS + j) * OCH + m) * ICH;
  v2f bv0 = { wp[2 * hi + 0], wp[2 * hi + 1] };
  v2f bv1 = { wp[4 + 2 * hi + 0], wp[4 + 2 * hi + 1] };

  float partial = 0.0f;
  for (int bt = 0; bt < BTILES; ++bt) {
    int b0 = bt * 16;
    const float* xp = x + ((size_t)(b0 + m) * ROUTES + r) * ICH;
    v2f a0 = { xp[2 * hi + 0], xp[2 * hi + 1] };
    v2f a1 = { xp[4 + 2 * hi + 0], xp[4 + 2 * hi + 1] };
    v8f acc = {};
    acc = wmma_k4(a0, bv0, acc);
    acc = wmma_k4(a1, bv1, acc);
    #pragma unroll
    for (int g = 0; g < 8; ++g) {
      int M = g + 8 * hi;
      partial += acc[g] * v[((size_t)(b0 + M) * NCAPS + j) * OCH + m];
    }
  }
  // wave32 reduction
  #pragma unroll
  for (int off = 16; off > 0; off >>= 1)
    partial += __shfl_down(partial, off, 32);
  if (lane == 0) b[r * NCAPS + j] += partial * (1.0f / (float)BATCH);
}

extern "C" void kernel_launch(void* const* d_in, const int* in_sizes, int n_in,
                              void* d_out, int out_size, void* d_ws, size_t ws_size,
                              hipStream_t stream) {
  const float* x = (const float*)d_in[0];  // [512,1152,8]
  const float* W = (const float*)d_in[1];  // [1152,10,16,8]
  float* v_out = (float*)d_out;            // [512,10,16,(1)] = 81920 f32

  // Workspace layout (floats): b | c | s_partial[8][81920]  (~2.7 MB)
  float* ws = (float*)d_ws;
  float* b = ws;
  float* c = b + ROUTES * NCAPS;
  float* s_partial = c + ROUTES * NCAPS;

  init_b_kernel<<<(ROUTES * NCAPS + 255) / 256, 256, 0, stream>>>(b);

  const int NITER = 3;
  for (int it = 0; it < NITER; ++it) {
    softmax_routes<<<NCAPS, 256, 0, stream>>>(b, c);
    compute_s_partial<<<BTILES * NCAPS * RCHUNKS, 32, 0, stream>>>(
        x, W, c, s_partial);
    reduce_squash<<<(SVOL + 255) / 256, 256, 0, stream>>>(s_partial, v_out);
    if (it < NITER - 1) {
      agreement_update_b<<<ROUTES * NCAPS, 32, 0, stream>>>(x, W, v_out, b);
    }
  }
}